// Decoder_64553358459245
// MI455X (gfx1250) — compile-verified
//
#include <hip/hip_runtime.h>
#include <math.h>

// fp32 WMMA fragments for V_WMMA_F32_16X16X4_F32 (wave32):
//   A 16x4 f32  -> 2 VGPRs/lane, B 4x16 f32 -> 2 VGPRs/lane, C/D 16x16 -> 8 VGPRs/lane
typedef float v2f __attribute__((ext_vector_type(2)));
typedef float v8f __attribute__((ext_vector_type(8)));

#define WMMA_F32_16X16X4(A, Bf, C) \
    __builtin_amdgcn_wmma_f32_16x16x4_f32(false, (A), false, (Bf), (short)0, (C), false, false)

// Problem constants (fixed by the reference file).
// B=32, T=128, M=64, P=64, 4P=256

// ---------------------------------------------------------------------------
// Kernel 1: UH[b,t,n] = sum_m H[b,t,m] * U_d[n,m]
// A = H viewed as [4096 x 64], B[k][n] = U_d[n][k]  -> D [4096 x 64]
// grid.x = 256 row-tiles of 16; block = 128 threads = 4 waves; wave w owns
// column tile n0 = 16*w; each wave chains K=64/4 = 16 f32 WMMAs.
// ---------------------------------------------------------------------------
__global__ __launch_bounds__(128)
void uh_gemm_kernel(const float* __restrict__ Hm,
                    const float* __restrict__ Ud,
                    float* __restrict__ UH) {
    const int row0 = blockIdx.x * 16;
    const int wave = threadIdx.x >> 5;
    const int lane = threadIdx.x & 31;
    const int n0   = wave * 16;
    const int half = lane >> 4;   // 0: K={0,1}(+4k)  1: K={2,3}(+4k)
    const int l    = lane & 15;

    const float* arow = Hm + (size_t)(row0 + l) * 64;  // A row for this lane
    const float* brow = Ud + (size_t)(n0 + l) * 64;    // B col N=l is U_d row (n0+l)

    v8f acc = {0.f, 0.f, 0.f, 0.f, 0.f, 0.f, 0.f, 0.f};
#pragma unroll
    for (int k = 0; k < 16; ++k) {
        const int ko = 4 * k + 2 * half;
        v2f av = *(const v2f*)(arow + ko);
        v2f bv = *(const v2f*)(brow + ko);
        acc = WMMA_F32_16X16X4(av, bv, acc);
    }
    // D layout: lanes 0-15 -> M = v, N = l ; lanes 16-31 -> M = v+8, N = l
#pragma unroll
    for (int v = 0; v < 8; ++v) {
        UH[(size_t)(row0 + v + 8 * half) * 64 + n0 + l] = acc[v];
    }
}

// ---------------------------------------------------------------------------
// Kernel 2: persistent single-workgroup sequential decoder.
// 1024 threads = 32 wave32s on one WGP. 127 LSTM steps + final attend.
// State and scratch live in LDS; gate tiles alias attention scratch
// (disjoint lifetimes). The per-step h @ W_hh^T GEMM (32x64 * 64x256) is
// exactly 32 16x16 output tiles -> one WMMA chain per wave.
// ---------------------------------------------------------------------------
__global__ __launch_bounds__(1024, 1)
void decoder_kernel(const float* __restrict__ Hm,     // [32,128,64]
                    const float* __restrict__ dec,    // [32,128,1]
                    const float* __restrict__ d1,     // [1,32,64]
                    const float* __restrict__ s1,     // [1,32,64]
                    const float* __restrict__ W_d,    // [64,128]
                    const float* __restrict__ v_d,    // [1,64]
                    const float* __restrict__ wtW,    // [1,65]
                    const float* __restrict__ wtb,    // [1]
                    const float* __restrict__ W_ih,   // [256,1]
                    const float* __restrict__ W_hh,   // [256,64]
                    const float* __restrict__ b_ih,   // [256]
                    const float* __restrict__ b_hh,   // [256]
                    const float* __restrict__ UH,     // [32,128,64] (ws)
                    float* __restrict__ out) {        // [32,1,128]
    __shared__ float sh_h[32 * 64];      // 8 KB
    __shared__ float sh_c[32 * 64];      // 8 KB
    __shared__ float sh_scr[8192];       // 32 KB: a|beta|ctx  (aliased by gates)
    __shared__ float sh_vd[64];
    __shared__ float sh_yt[32];

    const int tid  = threadIdx.x;
    const int wave = tid >> 5;
    const int lane = tid & 31;
    const int half = lane >> 4;
    const int l    = lane & 15;

    float* aB    = sh_scr;               // [32][64]   q @ W_d^T
    float* beta  = sh_scr + 2048;        // [32][128]  logits -> softmax
    float* ctx   = sh_scr + 6144;        // [32][64]
    float* gates = sh_scr;               // [32][256]  aliases a|beta|ctx

    // init h = d_1[0], c = s_1[0]
    for (int idx = tid; idx < 2048; idx += 1024) {
        sh_h[idx] = d1[idx];
        sh_c[idx] = s1[idx];
    }
    if (tid < 64) sh_vd[tid] = v_d[tid];
    __syncthreads();

    for (int s = 0; s < 128; ++s) {
        // ---- phase 1: a[b][n] = sum_p h[b][p]*W_d[n][p] + c[b][p]*W_d[n][64+p]
#pragma unroll
        for (int i = 0; i < 2; ++i) {
            const int idx = tid + i * 1024;
            const int b = idx >> 6, n = idx & 63;
            const float* wrow = W_d + (size_t)n * 128;
            float acc = 0.f;
            for (int p = 0; p < 64; ++p)
                acc = fmaf(sh_h[b * 64 + p], wrow[p],
                      fmaf(sh_c[b * 64 + p], wrow[64 + p], acc));
            aB[idx] = acc;
        }
        __syncthreads();

        // ---- phase 2: logits[b][t] = sum_m tanh(UH[b,t,m] + a[b,m]) * v_d[m]
#pragma unroll
        for (int i = 0; i < 4; ++i) {
            const int idx = tid + i * 1024;
            const int b = idx >> 7, t = idx & 127;
            const float* uh = UH + (size_t)(b * 128 + t) * 64;
            const float* av = aB + b * 64;
            float dot = 0.f;
            for (int m = 0; m < 64; ++m)
                dot = fmaf(tanhf(uh[m] + av[m]), sh_vd[m], dot);
            beta[idx] = dot;
        }
        __syncthreads();

        // ---- phase 3: softmax over T per batch row; wave w owns row b = w
        {
            const int b = wave;
            float v0[4];
            float mx = -INFINITY;
#pragma unroll
            for (int i = 0; i < 4; ++i) {
                v0[i] = beta[b * 128 + lane + 32 * i];
                mx = fmaxf(mx, v0[i]);
            }
            for (int off = 16; off; off >>= 1) mx = fmaxf(mx, __shfl_xor(mx, off, 32));
            float sum = 0.f;
#pragma unroll
            for (int i = 0; i < 4; ++i) { v0[i] = expf(v0[i] - mx); sum += v0[i]; }
            for (int off = 16; off; off >>= 1) sum += __shfl_xor(sum, off, 32);
            const float inv = 1.f / sum;
#pragma unroll
            for (int i = 0; i < 4; ++i) beta[b * 128 + lane + 32 * i] = v0[i] * inv;
        }
        __syncthreads();

        // ---- phase 4: ctx[b][m] = sum_t beta[b][t] * H[b][t][m]
#pragma unroll
        for (int i = 0; i < 2; ++i) {
            const int idx = tid + i * 1024;
            const int b = idx >> 6, m = idx & 63;
            const float* hb = Hm + (size_t)b * 128 * 64 + m;
            const float* bb = beta + b * 128;
            float acc = 0.f;
            for (int t = 0; t < 128; ++t) acc = fmaf(bb[t], hb[t * 64], acc);
            ctx[idx] = acc;
        }
        __syncthreads();

        if (s < 127) {
            // ---- phase 5: y_tilde[b] (tiny; 32 lanes)
            if (tid < 32) {
                const int b = tid;
                float y = wtW[0] * dec[b * 128 + s];
                for (int m = 0; m < 64; ++m)
                    y = fmaf(wtW[1 + m], ctx[b * 64 + m], y);
                sh_yt[b] = y + wtb[0];
            }
            __syncthreads();   // a/beta/ctx now dead -> gates may alias

            // ---- phase 6: gates_wm = h @ W_hh^T via f32 WMMA (one tile/wave)
            {
                const int b0 = (wave & 1) * 16;   // batch-tile
                const int n0 = (wave >> 1) * 16;  // gate-col tile
                const float* arow = sh_h + (size_t)(b0 + l) * 64;
                const float* brow = W_hh + (size_t)(n0 + l) * 64;  // B[k][n]=W_hh[n][k]
                v8f acc = {0.f, 0.f, 0.f, 0.f, 0.f, 0.f, 0.f, 0.f};
#pragma unroll
                for (int k = 0; k < 16; ++k) {
                    const int ko = 4 * k + 2 * half;
                    v2f av = *(const v2f*)(arow + ko);
                    v2f bv = *(const v2f*)(brow + ko);
                    acc = WMMA_F32_16X16X4(av, bv, acc);
                }
#pragma unroll
                for (int v = 0; v < 8; ++v)
                    gates[(b0 + v + 8 * half) * 256 + n0 + l] = acc[v];
            }
            __syncthreads();

            // ---- phase 7: LSTM nonlinearity, update h/c
#pragma unroll
            for (int i = 0; i < 2; ++i) {
                const int idx = tid + i * 1024;
                const int b = idx >> 6, p = idx & 63;
                const float yb = sh_yt[b];
                const float gi = gates[b * 256 + p]       + yb * W_ih[p]       + b_ih[p]       + b_hh[p];
                const float gf = gates[b * 256 + 64 + p]  + yb * W_ih[64 + p]  + b_ih[64 + p]  + b_hh[64 + p];
                const float gg = gates[b * 256 + 128 + p] + yb * W_ih[128 + p] + b_ih[128 + p] + b_hh[128 + p];
                const float go = gates[b * 256 + 192 + p] + yb * W_ih[192 + p] + b_ih[192 + p] + b_hh[192 + p];
                const float si = 1.f / (1.f + expf(-gi));
                const float sf = 1.f / (1.f + expf(-gf));
                const float so = 1.f / (1.f + expf(-go));
                const float cn = sf * sh_c[idx] + si * tanhf(gg);
                sh_c[idx] = cn;
                sh_h[idx] = so * tanhf(cn);
            }
            __syncthreads();
        } else {
            // ---- final: out[b][0][:] = cat(h[b], ctx[b])
#pragma unroll
            for (int i = 0; i < 4; ++i) {
                const int idx = tid + i * 1024;
                const int b = idx >> 7, j = idx & 127;
                out[idx] = (j < 64) ? sh_h[b * 64 + j] : ctx[b * 64 + (j - 64)];
            }
        }
    }
}

// ---------------------------------------------------------------------------
extern "C" void kernel_launch(void* const* d_in, const int* in_sizes, int n_in,
                              void* d_out, int out_size, void* d_ws, size_t ws_size,
                              hipStream_t stream) {
    const float* H    = (const float*)d_in[0];
    const float* dec  = (const float*)d_in[1];
    const float* d1   = (const float*)d_in[2];
    const float* s1   = (const float*)d_in[3];
    const float* W_d  = (const float*)d_in[4];
    const float* U_d  = (const float*)d_in[5];
    const float* v_d  = (const float*)d_in[6];
    const float* wtW  = (const float*)d_in[7];
    const float* wtb  = (const float*)d_in[8];
    const float* W_ih = (const float*)d_in[9];
    const float* W_hh = (const float*)d_in[10];
    const float* b_ih = (const float*)d_in[11];
    const float* b_hh = (const float*)d_in[12];
    (void)in_sizes; (void)n_in; (void)out_size; (void)ws_size;

    float* UH  = (float*)d_ws;          // 4096*64 f32 = 1 MB scratch (L2-resident)
    float* out = (float*)d_out;

    // Phase 1: grid-parallel fp32-WMMA GEMM for UH (hoisted out of the scan).
    uh_gemm_kernel<<<256, 128, 0, stream>>>(H, U_d, UH);
    // Phase 2: persistent single-WGP sequential decoder (127 steps + final).
    decoder_kernel<<<1, 1024, 0, stream>>>(H, dec, d1, s1, W_d, v_d, wtW, wtb,
                                           W_ih, W_hh, b_ih, b_hh, UH, out);
}